// EdgeAwareGNNLayer_25881472926479
// MI455X (gfx1250) — compile-verified
//
#include <hip/hip_runtime.h>
#include <math.h>

#define HDIM 128
#define BP 32       // B*P
#define NW 64
#define NT 64
#define MROWS 2048  // BP*64

typedef float v2f __attribute__((ext_vector_type(2)));
typedef float v8f __attribute__((ext_vector_type(8)));

// ---------------- tiny projection: out[m,n] = sum_{k<4} A[m,k]*Wt[n,k] + b[n]
__global__ void k_smallproj(const float* __restrict__ A, const float* __restrict__ Wt,
                            const float* __restrict__ b, float* __restrict__ out, int M) {
  int idx = blockIdx.x * blockDim.x + threadIdx.x;
  if (idx >= M * HDIM) return;
  int m = idx / HDIM, n = idx % HDIM;
  const float* a = A + (size_t)m * 4;
  const float* w = Wt + (size_t)n * 4;
  out[idx] = a[0]*w[0] + a[1]*w[1] + a[2]*w[2] + a[3]*w[3] + b[n];
}

// ---------------- u[h] = sum_k ec[h,k]*ep_w[k];  vb[h] = sum_k ec[h,k]*ep_b[k] + eu_b[h]
__global__ void k_prep_uv(const float* __restrict__ eu_w, const float* __restrict__ eu_b,
                          const float* __restrict__ ep_w, const float* __restrict__ ep_b,
                          float* __restrict__ u, float* __restrict__ vb) {
  int h = threadIdx.x;
  float su = 0.0f, sv = 0.0f;
  for (int k = 0; k < HDIM; ++k) {
    float ec = eu_w[(size_t)h * 384 + 256 + k];
    su += ec * ep_w[k];
    sv += ec * ep_b[k];
  }
  u[h]  = su;
  vb[h] = sv + eu_b[h];
}

// ---------------- generic WMMA f32 GEMM: C[m,n] = sum_k A[m,k]*Wt[n,k] + bscale*bias[n]
// one 16x16 tile per wave, 4 waves per block; grid = (N/16, M/64)
__global__ void k_gemm_wmma(const float* __restrict__ A, int lda,
                            const float* __restrict__ Wt, int ldw,
                            const float* __restrict__ bias, float bscale,
                            float* __restrict__ C, int ldc, int K) {
  int lane = threadIdx.x & 31;
  int wave = threadIdx.x >> 5;
  int half = lane >> 4;     // 0: K 0..1, 1: K 2..3
  int l    = lane & 15;     // row of A frag / col of B frag
  int kof  = half * 2;
  int N0 = blockIdx.x * 16;
  int M0 = (blockIdx.y * 4 + wave) * 16;

  const float* arow = A  + (size_t)(M0 + l) * lda + kof;  // A[m, k]
  const float* brow = Wt + (size_t)(N0 + l) * ldw + kof;  // B[k, n] = Wt[n, k]
  v8f acc = {};
  for (int k = 0; k < K; k += 4) {
    v2f a = *(const v2f*)(arow + k);
    v2f b = *(const v2f*)(brow + k);
    acc = __builtin_amdgcn_wmma_f32_16x16x4_f32(false, a, false, b, (short)0, acc,
                                                false, false);
  }
  float bv = bias ? bscale * bias[N0 + l] : 0.0f;
  // C/D layout: VGPR j -> row M0 + j + 8*half, col N0 + l
#pragma unroll
  for (int j = 0; j < 8; ++j) {
    C[(size_t)(M0 + j + half * 8) * ldc + N0 + l] = acc[j] + bv;
  }
}

// ---------------- edge kernel: one block per (bp*64 + w) row; writes edge_h_new and Sw (t-sum)
__global__ void k_edge(const float* __restrict__ Aw, const float* __restrict__ Bt,
                       const float* __restrict__ u, const float* __restrict__ e,
                       float* __restrict__ edge_out, float* __restrict__ Sw) {
  int blk = blockIdx.x;           // bp*64 + w
  int bp  = blk >> 6;
  int tz  = threadIdx.x >> 5;     // wave id = t lane (0..7)
  int h4  = threadIdx.x & 31;     // float4 index across H

  float4 aw = ((const float4*)(Aw + (size_t)blk * HDIM))[h4];
  float4 uu = ((const float4*)u)[h4];
  const float* ebase = e + (size_t)blk * NT;

  float ax = 0.0f, ay = 0.0f, az = 0.0f, awc = 0.0f;
  for (int tt = 0; tt < 8; ++tt) {
    int t = tt * 8 + tz;
    float ev = ebase[t];
    float4 bt = ((const float4*)(Bt + (size_t)(bp * NT + t) * HDIM))[h4];
    float4 v;
    v.x = tanhf(aw.x + bt.x + ev * uu.x);
    v.y = tanhf(aw.y + bt.y + ev * uu.y);
    v.z = tanhf(aw.z + bt.z + ev * uu.z);
    v.w = tanhf(aw.w + bt.w + ev * uu.w);
    ((float4*)(edge_out + ((size_t)blk * NT + t) * HDIM))[h4] = v;
    ax += v.x; ay += v.y; az += v.z; awc += v.w;
  }
  __shared__ float red[8 * HDIM];
  float* rp = red + tz * HDIM + h4 * 4;
  rp[0] = ax; rp[1] = ay; rp[2] = az; rp[3] = awc;
  __syncthreads();
  if (threadIdx.x < 32) {
    int hb = threadIdx.x;
    float sx = 0, sy = 0, sz = 0, sw = 0;
    for (int z = 0; z < 8; ++z) {
      const float* q = red + z * HDIM + hb * 4;
      sx += q[0]; sy += q[1]; sz += q[2]; sw += q[3];
    }
    float4 s; s.x = sx; s.y = sy; s.z = sz; s.w = sw;
    ((float4*)(Sw + (size_t)blk * HDIM))[hb] = s;
  }
}

// ---------------- St[bp,t,h] = sum_w edge[bp,w,t,h]
__global__ void k_colsum(const float* __restrict__ edge_out, float* __restrict__ St) {
  int idx = blockIdx.x * blockDim.x + threadIdx.x;   // (bp*64+t)*128+h
  if (idx >= MROWS * HDIM) return;
  int h = idx % HDIM;
  int r = idx / HDIM;
  int bp = r >> 6, t = r & 63;
  const float* base = edge_out + ((size_t)bp * 4096 + t) * HDIM + h;
  float s = 0.0f;
  for (int w = 0; w < NW; ++w) s += base[(size_t)w * 64 * HDIM];
  St[idx] = s;
}

// ---------------- dst[bp,h] = sum_{i<64} src[bp,i,h]
__global__ void k_rowsum64(const float* __restrict__ src, float* __restrict__ dst) {
  int idx = blockIdx.x * blockDim.x + threadIdx.x;
  if (idx >= BP * HDIM) return;
  int bp = idx / HDIM, h = idx % HDIM;
  const float* base = src + (size_t)bp * 64 * HDIM + h;
  float s = 0.0f;
  for (int i = 0; i < 64; ++i) s += base[(size_t)i * HDIM];
  dst[idx] = s;
}

// ---------------- X[m, 0:128]=scale*ent_h[m]; [128:256]=peer_sum[bp]; [256:384]=edge_sum[m]
__global__ void k_buildX(const float* __restrict__ ent_h, const float* __restrict__ peer_sum,
                         const float* __restrict__ edge_sum, float scale,
                         float* __restrict__ X) {
  int idx = blockIdx.x * blockDim.x + threadIdx.x;
  if (idx >= MROWS * 384) return;
  int m = idx / 384, k = idx % 384;
  int bp = m >> 6;
  float v;
  if (k < 128)      v = scale * ent_h[(size_t)m * HDIM + k];
  else if (k < 256) v = peer_sum[(size_t)bp * HDIM + (k - 128)];
  else              v = edge_sum[(size_t)m * HDIM + (k - 256)];
  X[idx] = v;
}

// ---------------- GRU gates: out = (1-z)*n + z*h
__global__ void k_gru(const float* __restrict__ gi, const float* __restrict__ gh,
                      const float* __restrict__ hprev, float* __restrict__ out) {
  int idx = blockIdx.x * blockDim.x + threadIdx.x;
  if (idx >= MROWS * HDIM) return;
  int m = idx / HDIM, h = idx % HDIM;
  const float* gim = gi + (size_t)m * 384;
  const float* ghm = gh + (size_t)m * 384;
  float ir = gim[h], iz = gim[h + 128], in_ = gim[h + 256];
  float hr = ghm[h], hz = ghm[h + 128], hn  = ghm[h + 256];
  float r = 1.0f / (1.0f + expf(-(ir + hr)));
  float z = 1.0f / (1.0f + expf(-(iz + hz)));
  float n = tanhf(in_ + r * hn);
  out[idx] = (1.0f - z) * n + z * hprev[idx];
}

extern "C" void kernel_launch(void* const* d_in, const int* in_sizes, int n_in,
                              void* d_out, int out_size, void* d_ws, size_t ws_size,
                              hipStream_t stream) {
  (void)in_sizes; (void)n_in; (void)out_size; (void)ws_size;
  const float* wf     = (const float*)d_in[0];
  const float* tf     = (const float*)d_in[1];
  const float* ef     = (const float*)d_in[2];
  const float* wp_w   = (const float*)d_in[3];
  const float* wp_b   = (const float*)d_in[4];
  const float* tp_w   = (const float*)d_in[5];
  const float* tp_b   = (const float*)d_in[6];
  const float* ep_w   = (const float*)d_in[7];
  const float* ep_b   = (const float*)d_in[8];
  const float* eu_w   = (const float*)d_in[9];
  const float* eu_b   = (const float*)d_in[10];
  const float* wm_w   = (const float*)d_in[11];
  const float* wm_b   = (const float*)d_in[12];
  const float* tm_w   = (const float*)d_in[13];
  const float* tm_b   = (const float*)d_in[14];
  const float* w_ih_w = (const float*)d_in[15];
  const float* w_hh_w = (const float*)d_in[16];
  const float* b_ih_w = (const float*)d_in[17];
  const float* b_hh_w = (const float*)d_in[18];
  const float* w_ih_t = (const float*)d_in[19];
  const float* w_hh_t = (const float*)d_in[20];
  const float* b_ih_t = (const float*)d_in[21];
  const float* b_hh_t = (const float*)d_in[22];

  float* out      = (float*)d_out;
  float* out_wh   = out;              // weapon_h_new: 262144
  float* out_th   = out + 262144;     // target_h_new: 262144
  float* edge_out = out + 524288;     // edge_h_new:   16777216

  float* ws = (float*)d_ws;
  float* weapon_h = ws;            ws += 262144;
  float* target_h = ws;            ws += 262144;
  float* Aw       = ws;            ws += 262144;
  float* Bt       = ws;            ws += 262144;
  float* Sw       = ws;            ws += 262144;
  float* St       = ws;            ws += 262144;
  float* tsum     = ws;            ws += 4096;
  float* wsum     = ws;            ws += 4096;
  float* u_vec    = ws;            ws += 128;
  float* vb_vec   = ws;            ws += 128;
  float* Xw       = ws;            ws += 786432;
  float* Xt       = ws;            ws += 786432;
  float* msg_w    = ws;            ws += 262144;
  float* msg_t    = ws;            ws += 262144;
  float* gi_w     = ws;            ws += 786432;
  float* gh_w     = ws;            ws += 786432;
  float* gi_t     = ws;            ws += 786432;
  float* gh_t     = ws;            ws += 786432;

  const int TB = 256;
  dim3 b256(TB);

  auto gemm = [&](const float* A, int lda, const float* Wt, int ldw,
                  const float* bias, float bscale, float* C, int ldc, int N, int K) {
    dim3 g(N / 16, MROWS / 64);
    k_gemm_wmma<<<g, dim3(128), 0, stream>>>(A, lda, Wt, ldw, bias, bscale, C, ldc, K);
  };

  // 1. input projections (K=4)
  k_smallproj<<<dim3((MROWS * HDIM) / TB), b256, 0, stream>>>(wf, wp_w, wp_b, weapon_h, MROWS);
  k_smallproj<<<dim3((MROWS * HDIM) / TB), b256, 0, stream>>>(tf, tp_w, tp_b, target_h, MROWS);

  // 2. rank-1 edge collapse vectors
  k_prep_uv<<<dim3(1), dim3(HDIM), 0, stream>>>(eu_w, eu_b, ep_w, ep_b, u_vec, vb_vec);

  // 3. Aw = weapon_h @ ea.T ; Bt = target_h @ eb.T + vb  (ea/eb are column slices of eu_w)
  gemm(weapon_h, HDIM, eu_w,       384, nullptr, 0.0f, Aw, HDIM, HDIM, HDIM);
  gemm(target_h, HDIM, eu_w + 128, 384, vb_vec,  1.0f, Bt, HDIM, HDIM, HDIM);

  // 4. edge_h_new = tanh(Aw + Bt + e*u) ; fused t-sum -> Sw
  k_edge<<<dim3(MROWS), b256, 0, stream>>>(Aw, Bt, u_vec, ef, edge_out, Sw);

  // 5. w-sum -> St
  k_colsum<<<dim3((MROWS * HDIM) / TB), b256, 0, stream>>>(edge_out, St);

  // 6. peer sums
  k_rowsum64<<<dim3((BP * HDIM) / TB), b256, 0, stream>>>(target_h, tsum);
  k_rowsum64<<<dim3((BP * HDIM) / TB), b256, 0, stream>>>(weapon_h, wsum);

  // 7. fused message operands, K=384: [T*h | peer_sum | edge_sum]
  k_buildX<<<dim3((MROWS * 384) / TB), b256, 0, stream>>>(weapon_h, tsum, Sw, 64.0f, Xw);
  k_buildX<<<dim3((MROWS * 384) / TB), b256, 0, stream>>>(target_h, wsum, St, 64.0f, Xt);

  // 8. msgs = X @ wm_w.T + T*wm_b  (single K=384 GEMM each)
  gemm(Xw, 384, wm_w, 384, wm_b, 64.0f, msg_w, HDIM, HDIM, 384);
  gemm(Xt, 384, tm_w, 384, tm_b, 64.0f, msg_t, HDIM, HDIM, 384);

  // 9. GRU GEMMs (N=384)
  gemm(msg_w,    HDIM, w_ih_w, HDIM, b_ih_w, 1.0f, gi_w, 384, 384, HDIM);
  gemm(weapon_h, HDIM, w_hh_w, HDIM, b_hh_w, 1.0f, gh_w, 384, 384, HDIM);
  gemm(msg_t,    HDIM, w_ih_t, HDIM, b_ih_t, 1.0f, gi_t, 384, 384, HDIM);
  gemm(target_h, HDIM, w_hh_t, HDIM, b_hh_t, 1.0f, gh_t, 384, 384, HDIM);

  // 10. GRU gates -> outputs
  k_gru<<<dim3((MROWS * HDIM) / TB), b256, 0, stream>>>(gi_w, gh_w, weapon_h, out_wh);
  k_gru<<<dim3((MROWS * HDIM) / TB), b256, 0, stream>>>(gi_t, gh_t, target_h, out_th);
}